// ModelNew_25056839205150
// MI455X (gfx1250) — compile-verified
//
#include <hip/hip_runtime.h>

// Per-token-group (G=128) dynamic fp8-range quantization, MI455X (gfx1250).
//
// Memory-bound streaming kernel (~402 MB traffic, ~17 us floor @ 23.3 TB/s).
// CDNA5 data path: async global->LDS b128 copies (ASYNCcnt) with a per-wave
// double buffer, wave32 16-lane-cluster shuffle max, non-temporal stores.
//
// Stage = 256 threads * 8 bf16 = 2048 elems = 16 groups. A wave (32 lanes)
// covers 2 groups; each 16-lane cluster owns one group of 128.

#define QF_FP8_MAX   448.0f
#define QF_AMAX_FLR  1e-4f
#define QF_STAGE_ELEMS 2048u
#define QF_STAGE_BYTES 4096u

typedef float v4f __attribute__((ext_vector_type(4)));

__global__ __launch_bounds__(256) void quant_fp8_group_kernel(
    const unsigned short* __restrict__ x,   // bf16 bit patterns
    float* __restrict__ y,                  // [M*N] fp32
    float* __restrict__ scale,              // [M*Ng] fp32
    unsigned stages)
{
    __shared__ uint4 buf[2][256];           // 8 KB double buffer, 16 B/lane/stage

    const unsigned tid    = threadIdx.x;
    const unsigned stride = gridDim.x;
    const unsigned s0     = blockIdx.x;
    if (s0 >= stages) return;

    // ---- prologue: issue async copy of first stage slice (16 B per lane) ----
    // global_load_async_to_lds_b128  VDST(=LDS byte offset), VADDR(=u32 byte
    // offset), SADDR(=u64 base)  -- GVS mode, tracked by ASYNCcnt.
    // Generic LDS pointers carry the wave-relative LDS offset in their low
    // 32 bits (aperture encoding), so a truncating cast yields VDST.
    {
        unsigned lds = (unsigned)(size_t)&buf[0][tid];
        unsigned off = s0 * QF_STAGE_BYTES + tid * 16u;
        asm volatile("global_load_async_to_lds_b128 %0, %1, %2 th:TH_LOAD_NT"
                     :: "v"(lds), "v"(off), "s"(x) : "memory");
    }

    unsigned b = 0;
    for (unsigned s = s0; s < stages; s += stride) {
        // Only one async load outstanding at this point: wait for it.
        asm volatile("s_wait_asynccnt 0x0" ::: "memory");

        uint4 r = buf[b][tid];              // ds_load_b128 of our own slice

        // Prefetch next stage into the other buffer; overlaps compute+store.
        unsigned next = s + stride;
        if (next < stages) {
            unsigned lds = (unsigned)(size_t)&buf[b ^ 1][tid];
            unsigned off = next * QF_STAGE_BYTES + tid * 16u;
            asm volatile("global_load_async_to_lds_b128 %0, %1, %2 th:TH_LOAD_NT"
                         :: "v"(lds), "v"(off), "s"(x) : "memory");
        }

        // ---- bf16 -> f32 (shift into exponent position) + local amax ----
        float v[8];
        v[0] = __uint_as_float(r.x << 16);
        v[1] = __uint_as_float(r.x & 0xffff0000u);
        v[2] = __uint_as_float(r.y << 16);
        v[3] = __uint_as_float(r.y & 0xffff0000u);
        v[4] = __uint_as_float(r.z << 16);
        v[5] = __uint_as_float(r.z & 0xffff0000u);
        v[6] = __uint_as_float(r.w << 16);
        v[7] = __uint_as_float(r.w & 0xffff0000u);

        float amax = fabsf(v[0]);
        #pragma unroll
        for (int i = 1; i < 8; ++i) amax = fmaxf(amax, fabsf(v[i]));

        // ---- 16-lane cluster max: one group of 128 per cluster (wave32) ----
        amax = fmaxf(amax, __shfl_xor(amax, 1, 32));
        amax = fmaxf(amax, __shfl_xor(amax, 2, 32));
        amax = fmaxf(amax, __shfl_xor(amax, 4, 32));
        amax = fmaxf(amax, __shfl_xor(amax, 8, 32));

        float a   = fmaxf(amax, QF_AMAX_FLR);
        float sc  = a * (1.0f / QF_FP8_MAX);
        float inv = QF_FP8_MAX / a;

        // ---- quantize + clamp, 32 B of output per lane, non-temporal ----
        size_t base = (size_t)s * QF_STAGE_ELEMS + (size_t)tid * 8u;
        v4f o0, o1;
        o0.x = fminf(fmaxf(v[0] * inv, -QF_FP8_MAX), QF_FP8_MAX);
        o0.y = fminf(fmaxf(v[1] * inv, -QF_FP8_MAX), QF_FP8_MAX);
        o0.z = fminf(fmaxf(v[2] * inv, -QF_FP8_MAX), QF_FP8_MAX);
        o0.w = fminf(fmaxf(v[3] * inv, -QF_FP8_MAX), QF_FP8_MAX);
        o1.x = fminf(fmaxf(v[4] * inv, -QF_FP8_MAX), QF_FP8_MAX);
        o1.y = fminf(fmaxf(v[5] * inv, -QF_FP8_MAX), QF_FP8_MAX);
        o1.z = fminf(fmaxf(v[6] * inv, -QF_FP8_MAX), QF_FP8_MAX);
        o1.w = fminf(fmaxf(v[7] * inv, -QF_FP8_MAX), QF_FP8_MAX);
        __builtin_nontemporal_store(o0, (v4f*)(y + base));
        __builtin_nontemporal_store(o1, (v4f*)(y + base + 4));

        if ((tid & 15u) == 0u) {
            unsigned g = s * 16u + (tid >> 4);   // flat group id == scale index
            __builtin_nontemporal_store(sc, scale + g);
        }

        b ^= 1;
    }
}

extern "C" void kernel_launch(void* const* d_in, const int* in_sizes, int n_in,
                              void* d_out, int out_size, void* d_ws, size_t ws_size,
                              hipStream_t stream) {
    (void)n_in; (void)out_size; (void)d_ws; (void)ws_size;
    const unsigned short* x = (const unsigned short*)d_in[0];  // bf16
    float* y = (float*)d_out;
    unsigned total  = (unsigned)in_sizes[0];        // M*N
    float* scale    = y + total;                    // outputs concatenated
    unsigned stages = total / QF_STAGE_ELEMS;       // 16 groups per stage
    if (stages == 0) return;
    unsigned grid = stages < 4096u ? stages : 4096u;
    quant_fp8_group_kernel<<<grid, 256, 0, stream>>>(x, y, scale, stages);
}